// HDModel_12197707120653
// MI455X (gfx1250) — compile-verified
//
#include <hip/hip_runtime.h>
#include <math.h>

#define DHV   10000
#define NSAMP 4096
#define NC    10
#define EPSF  1e-8f

typedef float v2f __attribute__((ext_vector_type(2)));
typedef float v8f __attribute__((ext_vector_type(8)));

// ---------------------------------------------------------------------------
// Kernel 1: build_am = onehot[16,4096] @ X[4096,10000] via v_wmma_f32_16x16x4_f32
// One wave per 16-column tile of D. A tile is generated from labels (LDS),
// B tile = 4 rows x 16 cols of X (two half-wave-coalesced b32 loads / step).
// ---------------------------------------------------------------------------
__global__ void __launch_bounds__(32) build_am_wmma(
    const float* __restrict__ X, const int* __restrict__ labels,
    float* __restrict__ am0) {
  __shared__ int s_lab[NSAMP];
  const int lane = threadIdx.x;
  for (int i = lane; i < NSAMP; i += 32) s_lab[i] = labels[i];
  __syncthreads();

  const int d0   = blockIdx.x * 16;   // column tile [d0, d0+16)
  const int nn   = lane & 15;
  const int koff = (lane >> 4) << 1;  // 0 for lanes 0-15, 2 for lanes 16-31

  v8f c = {0.f, 0.f, 0.f, 0.f, 0.f, 0.f, 0.f, 0.f};
  const float* bbase = X + (size_t)d0 + nn;

  for (int k = 0; k < NSAMP; k += 4) {
    const int l0 = s_lab[k + koff];
    const int l1 = s_lab[k + koff + 1];
    v2f a;
    a.x = (l0 == nn) ? 1.0f : 0.0f;   // A[M=nn][K=koff]
    a.y = (l1 == nn) ? 1.0f : 0.0f;   // A[M=nn][K=koff+1]
    const float* bp = bbase + (size_t)(k + koff) * DHV;
    v2f b;
    b.x = bp[0];                      // B[K=koff  ][N=nn] = X[k+koff  ][d0+nn]
    b.y = bp[DHV];                    // B[K=koff+1][N=nn] = X[k+koff+1][d0+nn]
    c = __builtin_amdgcn_wmma_f32_16x16x4_f32(false, a, false, b, (short)0, c,
                                              false, false);
  }

  // C layout: VGPR g -> row g (lanes 0-15) / row 8+g (lanes 16-31), col = lane&15
  const bool low = lane < 16;
#pragma unroll
  for (int g = 0; g < 8; ++g) {
    const int cls = low ? g : (8 + g);
    if (cls < NC) am0[(size_t)cls * DHV + d0 + nn] = c[g];
  }
}

// ---------------------------------------------------------------------------
// Kernel 2: sequential retrain scan (one workgroup). The entire 10x10000 AM
// lives in VGPRs (10 classes x 20 column slots per thread, 512 threads,
// ~225 VGPR peak: hv streamed in chunks of 4 so it is never fully live).
// Class norms maintained incrementally; reductions deterministic
// (shuffle + fixed-order LDS stage, no float atomics). Next sample row is
// prefetched before the barriers so the serial loop hides its memory latency.
// ---------------------------------------------------------------------------
#define NT2    512
#define NSLOT  20
#define CH     4
#define NCHUNK (NSLOT / CH)
#define NWAVE  (NT2 / 32)

__global__ void __launch_bounds__(NT2, 1) retrain_scan(
    const float* __restrict__ X, const int* __restrict__ labels,
    float* __restrict__ am,        // in: am0, out: am_final  (d_out region)
    float* __restrict__ amnT,      // ws: normalized transposed AM [DHV][16]
    float* __restrict__ mist_out) {
  __shared__ float s_part[NWAVE][NC + 1];
  __shared__ float s_red[NC + 1];
  __shared__ float s_n2[NC];
  __shared__ int   s_ctrl[3];

  const int t    = threadIdx.x;
  const int lane = t & 31;
  const int wid  = t >> 5;

  // ---- load am0 into registers (thread t owns columns t + s*512)
  float amreg[NC][NSLOT];
#pragma unroll
  for (int c = 0; c < NC; ++c)
#pragma unroll
    for (int s = 0; s < NSLOT; ++s) {
      const int col = t + s * NT2;
      amreg[c][s] = (col < DHV) ? am[(size_t)c * DHV + col] : 0.0f;
    }

  // ---- initial class norms^2
  {
    float part[NC];
#pragma unroll
    for (int c = 0; c < NC; ++c) {
      float acc = 0.0f;
#pragma unroll
      for (int s = 0; s < NSLOT; ++s) acc = fmaf(amreg[c][s], amreg[c][s], acc);
#pragma unroll
      for (int off = 16; off >= 1; off >>= 1) acc += __shfl_xor(acc, off, 32);
      part[c] = acc;
    }
    if (lane == 0) {
#pragma unroll
      for (int c = 0; c < NC; ++c) s_part[wid][c] = part[c];
    }
  }
  __syncthreads();
  if (t < NC) {
    float acc = 0.0f;
#pragma unroll
    for (int w = 0; w < NWAVE; ++w) acc += s_part[w][t];
    s_n2[t] = acc;
  }
  __syncthreads();

  float mist = 0.0f;

  for (int n = 0; n < NSAMP; ++n) {
    const float* __restrict__ Xn = X + (size_t)n * DHV;

    // ---- partial dots (10 classes) + |hv|^2, hv streamed in chunks of 4
    float part[NC + 1];
#pragma unroll
    for (int v = 0; v < NC + 1; ++v) part[v] = 0.0f;
#pragma unroll
    for (int sc = 0; sc < NCHUNK; ++sc) {
      float hc[CH];
#pragma unroll
      for (int j = 0; j < CH; ++j) {
        const int col = t + (sc * CH + j) * NT2;
        hc[j] = (col < DHV) ? Xn[col] : 0.0f;
      }
#pragma unroll
      for (int j = 0; j < CH; ++j) {
        const int s = sc * CH + j;
        part[NC] = fmaf(hc[j], hc[j], part[NC]);
#pragma unroll
        for (int c = 0; c < NC; ++c)
          part[c] = fmaf(amreg[c][s], hc[j], part[c]);
      }
    }
#pragma unroll
    for (int v = 0; v < NC + 1; ++v) {
#pragma unroll
      for (int off = 16; off >= 1; off >>= 1)
        part[v] += __shfl_xor(part[v], off, 32);
    }
    if (lane == 0) {
#pragma unroll
      for (int v = 0; v < NC + 1; ++v) s_part[wid][v] = part[v];
    }

    // ---- prefetch next sample row (1 lane per 64B line) before the barriers
    if (n + 1 < NSAMP && (t & 15) == 0) {
      const float* __restrict__ Xp = X + (size_t)(n + 1) * DHV;
#pragma unroll
      for (int s = 0; s < NSLOT; ++s) {
        const int col = t + s * NT2;
        if (col < DHV) __builtin_prefetch(Xp + col, 0, 0);
      }
    }
    __syncthreads();

    if (t < NC + 1) {                 // deterministic cross-wave reduce
      float acc = 0.0f;
#pragma unroll
      for (int w = 0; w < NWAVE; ++w) acc += s_part[w][t];
      s_red[t] = acc;
    }
    __syncthreads();

    if (t == 0) {
      const float h2 = s_red[NC];
      const float nh = sqrtf(h2);
      int pred = 0;
      float best = -3.402823466e38f;
#pragma unroll
      for (int c = 0; c < NC; ++c) {
        const float sim = s_red[c] / fmaxf(sqrtf(s_n2[c]) * nh, EPSF);
        if (sim > best) { best = sim; pred = c; }   // first-max tie-break
      }
      const int label = labels[n];
      const int wrong = (pred != label) ? 1 : 0;
      if (wrong) {
        // ||am +- hv||^2 = ||am||^2 +- 2*(am.hv) + ||hv||^2 (dot already known)
        s_n2[pred]  += h2 - 2.0f * s_red[pred];
        s_n2[label] += h2 + 2.0f * s_red[label];
        mist += 1.0f;
      }
      s_ctrl[0] = pred; s_ctrl[1] = label; s_ctrl[2] = wrong;
    }
    __syncthreads();

    if (s_ctrl[2]) {
      const int pred  = s_ctrl[0];
      const int label = s_ctrl[1];
      float sgn[NC];
#pragma unroll
      for (int c = 0; c < NC; ++c)
        sgn[c] = (c == label ? 1.0f : 0.0f) - (c == pred ? 1.0f : 0.0f);
      // re-stream the row (L2/WGP$-hot) instead of keeping it live
#pragma unroll
      for (int sc = 0; sc < NCHUNK; ++sc) {
        float hc[CH];
#pragma unroll
        for (int j = 0; j < CH; ++j) {
          const int col = t + (sc * CH + j) * NT2;
          hc[j] = (col < DHV) ? Xn[col] : 0.0f;
        }
#pragma unroll
        for (int j = 0; j < CH; ++j) {
          const int s = sc * CH + j;
#pragma unroll
          for (int c = 0; c < NC; ++c)
            amreg[c][s] = fmaf(sgn[c], hc[j], amreg[c][s]);
        }
      }
    }
    __syncthreads();   // protect s_part/s_red/s_ctrl for next iteration
  }

  // ---- write final AM (d_out) and mistake count
#pragma unroll
  for (int c = 0; c < NC; ++c)
#pragma unroll
    for (int s = 0; s < NSLOT; ++s) {
      const int col = t + s * NT2;
      if (col < DHV) am[(size_t)c * DHV + col] = amreg[c][s];
    }
  if (t == 0) *mist_out = mist;

  // ---- final norms, then emit padded/normalized/transposed AM for predict
  {
    float part[NC];
#pragma unroll
    for (int c = 0; c < NC; ++c) {
      float acc = 0.0f;
#pragma unroll
      for (int s = 0; s < NSLOT; ++s) acc = fmaf(amreg[c][s], amreg[c][s], acc);
#pragma unroll
      for (int off = 16; off >= 1; off >>= 1) acc += __shfl_xor(acc, off, 32);
      part[c] = acc;
    }
    if (lane == 0) {
#pragma unroll
      for (int c = 0; c < NC; ++c) s_part[wid][c] = part[c];
    }
  }
  __syncthreads();
  if (t < NC) {
    float acc = 0.0f;
#pragma unroll
    for (int w = 0; w < NWAVE; ++w) acc += s_part[w][t];
    s_red[t] = acc;
  }
  __syncthreads();

  float rn[NC];
#pragma unroll
  for (int c = 0; c < NC; ++c) rn[c] = 1.0f / fmaxf(sqrtf(s_red[c]), EPSF);
#pragma unroll
  for (int s = 0; s < NSLOT; ++s) {
    const int col = t + s * NT2;
    if (col < DHV) {
#pragma unroll
      for (int c = 0; c < NC; ++c)
        amnT[(size_t)col * 16 + c] = amreg[c][s] * rn[c];
#pragma unroll
      for (int c = NC; c < 16; ++c) amnT[(size_t)col * 16 + c] = 0.0f;
    }
  }
}

// ---------------------------------------------------------------------------
// Kernel 3: predict. argmax_c( x . amn_c ) — the ||x|| factor is a positive
// per-row scalar and cannot change the argmax, so X is not normalized.
// One wave per 16-row tile, K-loop of v_wmma_f32_16x16x4_f32, shuffle argmax
// epilogue directly from the C-tile VGPR layout.
// ---------------------------------------------------------------------------
__global__ void __launch_bounds__(32) predict_wmma(
    const float* __restrict__ X, const float* __restrict__ amnT,
    float* __restrict__ preds) {
  const int lane = threadIdx.x;
  const int m0   = blockIdx.x * 16;
  const int nn   = lane & 15;
  const int koff = (lane >> 4) << 1;

  const float* arow = X + (size_t)(m0 + nn) * DHV + koff;   // A[M=nn][K=koff..]
  const float* bcol = amnT + (size_t)koff * 16 + nn;        // B[K=koff][N=nn]

  v8f c = {0.f, 0.f, 0.f, 0.f, 0.f, 0.f, 0.f, 0.f};
  for (int k = 0; k < DHV; k += 4) {
    if ((k & 255) == 0) __builtin_prefetch(arow + k + 256, 0, 0);
    v2f a;
    a.x = arow[k];
    a.y = arow[k + 1];
    v2f b;
    b.x = bcol[(size_t)k * 16];
    b.y = bcol[(size_t)k * 16 + 16];
    c = __builtin_amdgcn_wmma_f32_16x16x4_f32(false, a, false, b, (short)0, c,
                                              false, false);
  }

  // argmax over classes per row: VGPR g holds row g (lanes 0-15) and row 8+g
  // (lanes 16-31), col = lane&15; reduce each 16-lane half independently.
#pragma unroll
  for (int g = 0; g < 8; ++g) {
    float v = c[g];
    int idx = nn;
    if (idx >= NC) v = -3.402823466e38f;
#pragma unroll
    for (int off = 8; off >= 1; off >>= 1) {
      const float ov = __shfl_xor(v, off, 32);
      const int   oi = __shfl_xor(idx, off, 32);
      if (ov > v || (ov == v && oi < idx)) { v = ov; idx = oi; }
    }
    if (lane == 0)  preds[m0 + g]     = (float)idx;
    if (lane == 16) preds[m0 + 8 + g] = (float)idx;
  }
}

// ---------------------------------------------------------------------------
extern "C" void kernel_launch(void* const* d_in, const int* in_sizes, int n_in,
                              void* d_out, int out_size, void* d_ws,
                              size_t ws_size, hipStream_t stream) {
  const float* X      = (const float*)d_in[0];   // [4096,10000] f32
  const int*   labels = (const int*)d_in[1];     // [4096]

  float* out   = (float*)d_out;
  float* am    = out;                       // [10,10000] am_final
  float* preds = out + (size_t)NC * DHV;    // [4096]
  float* mist  = preds + NSAMP;             // [1]
  float* amnT  = (float*)d_ws;              // [10000,16] = 640 KB scratch

  build_am_wmma<<<DHV / 16, 32, 0, stream>>>(X, labels, am);
  retrain_scan<<<1, NT2, 0, stream>>>(X, labels, am, amnT, mist);
  predict_wmma<<<NSAMP / 16, 32, 0, stream>>>(X, amnT, preds);
}